// MaSIF_ligand_22651657519751
// MI455X (gfx1250) — compile-verified
//
#include <hip/hip_runtime.h>
#include <hip/hip_bf16.h>

// ---------------- problem constants (match reference) ----------------
#define PP 32       // patches
#define VV 200      // vertices
#define FF 5        // features
#define GG 80       // gaussians (T*R)
#define KK 16       // rotations
#define BB 16       // batch
#define NL 7
#define BIG_LEN (PP * VV * FF)          // 32000
#define XROW (BIG_LEN + 3 * PP * VV)    // 51200
#define EPSV 1e-5f
#define TWO_PI 6.283185307179586f
#define INV_TWO_PI 0.15915494309189535f

typedef float v2f __attribute__((ext_vector_type(2)));
typedef float v8f __attribute__((ext_vector_type(8)));

// =====================================================================
// Kernel 1: per (b,p) patch.
//   phase A: gaussian soft-bins  desc[k][f*G+g]  (v-outer / k-inner,
//            rho-term hoisted, 16 accumulator pairs in registers)
//   phase B: conv = max_k( desc_k @ W_conv[f] ) + b_conv, relu
//            -> WMMA f32 16x16x4, M = 16 rotations, N,K = 80
//   phase C: h = relu(conv_flat(400) @ W1 + b1) -> h_all[b][p][80]
// =====================================================================
__global__ __launch_bounds__(512)
void masif_k1(const float* __restrict__ x,
              const float* __restrict__ mu_rho, const float* __restrict__ sigma_rho,
              const float* __restrict__ mu_theta, const float* __restrict__ sigma_theta,
              const float* __restrict__ W_conv, const float* __restrict__ b_conv,
              const float* __restrict__ W1, const float* __restrict__ b1,
              float* __restrict__ h_all)
{
    const int bp = blockIdx.x;
    const int b = bp / PP;
    const int p = bp % PP;
    const int tid = threadIdx.x;

    __shared__ float s_rho[VV];
    __shared__ float s_theta[VV];
    __shared__ float s_mask[VV];
    __shared__ float s_feat[VV * FF];        // [v*F + f]
    __shared__ float s_desc[KK][FF * GG];    // [k][f*80+g]  25.6 KB
    __shared__ float s_conv[FF * GG];        // conv flat (400)
    __shared__ float s_hpart[FF][GG];        // split-K partials for W1 matvec

    const float* xb = x + (size_t)b * XROW;
    for (int i = tid; i < VV; i += blockDim.x) {
        s_rho[i]   = xb[BIG_LEN + 0 * PP * VV + p * VV + i];
        s_theta[i] = xb[BIG_LEN + 1 * PP * VV + p * VV + i];
        s_mask[i]  = xb[BIG_LEN + 2 * PP * VV + p * VV + i];
    }
    for (int i = tid; i < VV * FF; i += blockDim.x)
        s_feat[i] = xb[p * VV * FF + i];
    __syncthreads();

    // ---------------- phase A: gaussian descriptors ----------------
    if (tid < FF * GG) {
        const int f = tid / GG;
        const int g = tid % GG;
        const float mr  = mu_rho[f * GG + g];
        const float sr  = sigma_rho[f * GG + g];
        const float isr = 1.0f / (sr * sr + EPSV);
        const float mt  = mu_theta[f * GG + g];
        const float st  = sigma_theta[f * GG + g];
        const float ist = 1.0f / (st * st + EPSV);

        float numer[KK], denom[KK];
        #pragma unroll
        for (int k = 0; k < KK; ++k) { numer[k] = 0.0f; denom[k] = 0.0f; }

        for (int v = 0; v < VV; ++v) {
            const float rv = s_rho[v];
            const float tv = s_theta[v];
            const float mv = s_mask[v];
            const float fv = s_feat[v * FF + f];
            const float dr = rv - mr;
            const float ar = dr * dr * isr;          // rho term: hoisted over k
            #pragma unroll
            for (int k = 0; k < KK; ++k) {
                float th = tv + (TWO_PI / (float)KK) * (float)k;
                th -= floorf(th * INV_TWO_PI) * TWO_PI;   // jnp.mod (args >= 0)
                const float dt = th - mt;
                const float w  = __expf(-(ar + dt * dt * ist)) * mv;
                numer[k] = fmaf(w, fv, numer[k]);
                denom[k] += w;
            }
        }
        #pragma unroll
        for (int k = 0; k < KK; ++k)
            s_desc[k][f * GG + g] = numer[k] / (denom[k] + EPSV);
    }
    __syncthreads();

    // ---------------- phase B: conv GEMM via WMMA f32 16x16x4 ----------------
    // 25 tile-jobs: f in [0,5) x ntile in [0,5).  M = rotations (16).
    {
        const int wave = tid >> 5;          // 16 waves
        const int lane = tid & 31;
        const int half = lane >> 4;         // 0: lanes 0-15, 1: lanes 16-31
        const int l15  = lane & 15;
        for (int job = wave; job < 25; job += 16) {
            const int f  = job / 5;
            const int n0 = (job % 5) * 16;
            v8f c = {};
            const float* Wf = W_conv + f * GG * GG;   // [g*80 + h]
            for (int k0 = 0; k0 < GG; k0 += 4) {
                const int ka = k0 + 2 * half;
                v2f a, bfr;
                a.x   = s_desc[l15][f * GG + ka];
                a.y   = s_desc[l15][f * GG + ka + 1];
                bfr.x = Wf[ka * GG + n0 + l15];
                bfr.y = Wf[(ka + 1) * GG + n0 + l15];
                c = __builtin_amdgcn_wmma_f32_16x16x4_f32(
                        false, a, false, bfr, (short)0, c, false, false);
            }
            // C layout: VGPR r, lanes<16 -> M=r, lanes>=16 -> M=r+8; N = n0+l15.
            // rotation-max == max over M rows.
            float m8 = c[0];
            m8 = fmaxf(m8, c[1]); m8 = fmaxf(m8, c[2]); m8 = fmaxf(m8, c[3]);
            m8 = fmaxf(m8, c[4]); m8 = fmaxf(m8, c[5]); m8 = fmaxf(m8, c[6]);
            m8 = fmaxf(m8, c[7]);
            float other = __shfl_xor(m8, 16, 32);
            float mall  = fmaxf(m8, other);
            if (half == 0) {
                float vconv = mall + b_conv[f * GG + n0 + l15];
                s_conv[f * GG + n0 + l15] = fmaxf(vconv, 0.0f);
            }
        }
    }
    __syncthreads();

    // ---------------- phase C: h = relu(conv_flat @ W1 + b1) ----------------
    // split-K over 5 chunks of 80: 400 threads produce partials, 80 reduce.
    if (tid < FF * GG) {
        const int o  = tid % GG;
        const int ch = tid / GG;           // 0..4
        const int i0 = ch * GG;
        float acc = 0.0f;
        for (int i = 0; i < GG; ++i)
            acc = fmaf(s_conv[i0 + i], W1[(i0 + i) * GG + o], acc);
        s_hpart[ch][o] = acc;
    }
    __syncthreads();
    if (tid < GG) {
        float acc = b1[tid];
        #pragma unroll
        for (int ch = 0; ch < FF; ++ch) acc += s_hpart[ch][tid];
        h_all[(size_t)bp * GG + tid] = fmaxf(acc, 0.0f);
    }
}

// =====================================================================
// Kernel 2: per batch b.
//   cov = h^T h / P  (80x80x32 GEMM via WMMA f32 16x16x4)
//   h2  = relu(cov_flat @ W2 + b2)     (scalar split-K)
//   out = softmax(h2 @ W3 + b3)
// =====================================================================
__global__ __launch_bounds__(512)
void masif_k2(const float* __restrict__ h_all,
              const float* __restrict__ W2, const float* __restrict__ b2,
              const float* __restrict__ W3, const float* __restrict__ b3,
              float* __restrict__ out)
{
    const int b = blockIdx.x;
    const int tid = threadIdx.x;

    __shared__ float s_h[PP][GG];        // 32x80
    __shared__ float s_cov[GG * GG];     // 6400
    __shared__ float s_part[8][64];
    __shared__ float s_h2[64];
    __shared__ float s_logit[NL];

    for (int i = tid; i < PP * GG; i += blockDim.x)
        s_h[i / GG][i % GG] = h_all[(size_t)b * PP * GG + i];
    __syncthreads();

    // ------------- cov via WMMA: M=80, N=80, K=32 -------------
    {
        const int wave = tid >> 5;
        const int lane = tid & 31;
        const int half = lane >> 4;
        const int l15  = lane & 15;
        for (int job = wave; job < 25; job += 16) {
            const int m0 = (job / 5) * 16;
            const int n0 = (job % 5) * 16;
            v8f c = {};
            for (int k0 = 0; k0 < PP; k0 += 4) {
                const int ka = k0 + 2 * half;
                v2f a, bfr;
                a.x   = s_h[ka][m0 + l15];       // A[m,k] = h[k][m]
                a.y   = s_h[ka + 1][m0 + l15];
                bfr.x = s_h[ka][n0 + l15];       // B[k,n] = h[k][n]
                bfr.y = s_h[ka + 1][n0 + l15];
                c = __builtin_amdgcn_wmma_f32_16x16x4_f32(
                        false, a, false, bfr, (short)0, c, false, false);
            }
            const float invP = 1.0f / (float)PP;
            #pragma unroll
            for (int r = 0; r < 8; ++r) {
                const int row = m0 + r + 8 * half;
                s_cov[row * GG + n0 + l15] = c[r] * invP;
            }
        }
    }
    __syncthreads();

    // ------------- h2 = relu(cov_flat(6400) @ W2 + b2), 64 outs -------------
    {
        const int o = tid & 63;          // output
        const int chunk = tid >> 6;      // 0..7 split-K
        float acc = 0.0f;
        const int i0 = chunk * 800;
        for (int i = 0; i < 800; ++i)
            acc = fmaf(s_cov[i0 + i], W2[(size_t)(i0 + i) * 64 + o], acc);
        s_part[chunk][o] = acc;
    }
    __syncthreads();
    if (tid < 64) {
        float acc = b2[tid];
        #pragma unroll
        for (int cch = 0; cch < 8; ++cch) acc += s_part[cch][tid];
        s_h2[tid] = fmaxf(acc, 0.0f);
    }
    __syncthreads();

    // ------------- logits + softmax -------------
    if (tid < NL) {
        float acc = b3[tid];
        for (int j = 0; j < 64; ++j)
            acc = fmaf(s_h2[j], W3[j * NL + tid], acc);
        s_logit[tid] = acc;
    }
    __syncthreads();
    if (tid < NL) {
        float m = s_logit[0];
        for (int j = 1; j < NL; ++j) m = fmaxf(m, s_logit[j]);
        float sum = 0.0f;
        for (int j = 0; j < NL; ++j) sum += __expf(s_logit[j] - m);
        out[b * NL + tid] = __expf(s_logit[tid] - m) / sum;
    }
}

extern "C" void kernel_launch(void* const* d_in, const int* in_sizes, int n_in,
                              void* d_out, int out_size, void* d_ws, size_t ws_size,
                              hipStream_t stream) {
    const float* x           = (const float*)d_in[0];
    const float* mu_rho      = (const float*)d_in[1];
    const float* sigma_rho   = (const float*)d_in[2];
    const float* mu_theta    = (const float*)d_in[3];
    const float* sigma_theta = (const float*)d_in[4];
    const float* W_conv      = (const float*)d_in[5];
    const float* b_conv      = (const float*)d_in[6];
    const float* W1          = (const float*)d_in[7];
    const float* b1          = (const float*)d_in[8];
    const float* W2          = (const float*)d_in[9];
    const float* b2          = (const float*)d_in[10];
    const float* W3          = (const float*)d_in[11];
    const float* b3          = (const float*)d_in[12];
    float* out = (float*)d_out;

    float* h_all = (float*)d_ws;   // B*P*G floats = 160 KB

    masif_k1<<<dim3(BB * PP), dim3(512), 0, stream>>>(
        x, mu_rho, sigma_rho, mu_theta, sigma_theta,
        W_conv, b_conv, W1, b1, h_all);

    masif_k2<<<dim3(BB), dim3(512), 0, stream>>>(
        h_all, W2, b2, W3, b3, out);
}